// DistributionLoss_80900003987548
// MI455X (gfx1250) — compile-verified
//
#include <hip/hip_runtime.h>
#include <stdint.h>

// Problem constants (from reference): B=16, C=3, H=W=512, K=7, PAD=3
#define IMG      512
#define NPIX     (512 * 512)
#define NCH      3
#define TILE     32          // output tile
#define HALO     3
#define REG      38          // TILE + 2*HALO
#define TSTRIDE  40          // padded row stride for raw/t/t2 planes
#define HROWS    40          // 38 valid + 2 zero rows for WMMA K-chunk overread
#define HSTRIDE  48          // rows k and k+2 hit disjoint LDS bank groups
#define NDIVF    147.0f      // C*K*K fixed divisor (zero padding counts)
#define NTOTAL   4194304.0f  // 16*512*512 elements in the mean
#define NTILES   4096        // 16 batch * 16x16 tiles
#define TPB      4           // tiles per persistent block
#define NBLOCKS  (NTILES / TPB)

// ---------------- CDNA5 async global->LDS path (guarded) ----------------
#if defined(__has_builtin)
#if __has_builtin(__builtin_amdgcn_global_load_async_to_lds_b32)
#define ASYNC_LDS 1
#endif
#endif
#ifndef ASYNC_LDS
#define ASYNC_LDS 0
#endif

#if ASYNC_LDS
#if __has_builtin(__builtin_amdgcn_s_wait_asynccnt)
#define WAIT_ASYNC() __builtin_amdgcn_s_wait_asynccnt(0)
#else
#define WAIT_ASYNC() asm volatile("s_wait_asynccnt 0" ::: "memory")
#endif
// Builtin signature (from clang diagnostic): first param is '__device__ int *'
// (address_space(1) int*, non-const); LDS param analogously '__shared__ int *'.
typedef __attribute__((address_space(1))) int g_int_t;
typedef __attribute__((address_space(3))) int l_int_t;
// int-route cast: strips const, changes pointee type, and retargets the
// address space (generic LDS pointer's low 32 bits are the LDS offset).
#define AS_G(p) ((g_int_t*)(uintptr_t)(p))
#define AS_L(p) ((l_int_t*)(uintptr_t)(p))
#else
#define WAIT_ASYNC() ((void)0)
#endif

typedef __attribute__((ext_vector_type(2))) float v2f;
typedef __attribute__((ext_vector_type(8))) float v8f;

// Stage one tile's 38x38 region (3 channels x 2 images) into LDS raw planes.
// Async path: LDS-direct DMA tracked by ASYNCcnt. OOB pixels: LDS zeros.
__device__ __forceinline__ void stage_tile(
    const float* __restrict__ pred, const float* __restrict__ tgt,
    int tile, float (*buf)[REG * TSTRIDE], int tid)
{
    const int bx = tile & 15, by = (tile >> 4) & 15, bz = tile >> 8;
    const int x0 = bx * TILE, y0 = by * TILE;
    const size_t imgBase = (size_t)bz * NCH * NPIX;
    for (int idx = tid; idx < REG * REG; idx += 128) {
        const int r = idx / REG, c = idx - r * REG;
        const int gy = y0 - HALO + r, gx = x0 - HALO + c;
        const int l = r * TSTRIDE + c;
        if ((unsigned)gy < (unsigned)IMG && (unsigned)gx < (unsigned)IMG) {
            const size_t o = imgBase + (size_t)gy * IMG + gx;
#if ASYNC_LDS
#pragma unroll
            for (int ch = 0; ch < NCH; ++ch) {
                __builtin_amdgcn_global_load_async_to_lds_b32(
                    AS_G(pred + o + (size_t)ch * NPIX), AS_L(&buf[ch][l]), 0, 0);
                __builtin_amdgcn_global_load_async_to_lds_b32(
                    AS_G(tgt + o + (size_t)ch * NPIX), AS_L(&buf[NCH + ch][l]), 0, 0);
            }
#else
#pragma unroll
            for (int ch = 0; ch < NCH; ++ch) {
                buf[ch][l]       = pred[o + (size_t)ch * NPIX];
                buf[NCH + ch][l] = tgt [o + (size_t)ch * NPIX];
            }
#endif
        } else {
#pragma unroll
            for (int p = 0; p < 2 * NCH; ++p) buf[p][l] = 0.0f;
        }
    }
}

__global__ __launch_bounds__(128) void local_std_smoothl1_kernel(
    const float* __restrict__ pred, const float* __restrict__ tgt,
    float* __restrict__ partial)
{
    __shared__ float raw[2][2 * NCH][REG * TSTRIDE];             // double-buffered channels
    __shared__ float tP[REG * TSTRIDE], qP[REG * TSTRIDE];       // pred: sum_c x, sum_c x^2
    __shared__ float tM[REG * TSTRIDE], qM[REG * TSTRIDE];       // target
    __shared__ float hsP[HROWS * HSTRIDE], hqP[HROWS * HSTRIDE]; // horizontal 7-sums
    __shared__ float hsM[HROWS * HSTRIDE], hqM[HROWS * HSTRIDE];
    __shared__ float red[128];

    const int tid  = threadIdx.x;
    const int lane = tid & 31;
    const int wave = tid >> 5;
    const int r0 = (wave >> 1) * 16;   // wave -> 16x16 tile of the 32x32 block tile
    const int c0 = (wave & 1) * 16;
    const int hi = lane >> 4;          // half-wave select
    const int m  = lane & 15;          // A row / B column N

    const int baseTile = (int)blockIdx.x * TPB;

    __builtin_prefetch(pred, 0, 1);    // global_prefetch_b8 path
    __builtin_prefetch(tgt, 0, 1);

    // prologue: start DMA for first tile
    stage_tile(pred, tgt, baseTile, raw[0], tid);

    for (int i = 0; i < TPB; ++i) {
        const int cb = i & 1;
        WAIT_ASYNC();                  // my async loads for buffer cb landed
        __syncthreads();               // everyone's landed (fallback: dscnt via barrier)

        // overlap: kick off next tile's DMA into the other buffer
        if (i + 1 < TPB) stage_tile(pred, tgt, baseTile + i + 1, raw[cb ^ 1], tid);

        // -------- channel fold: t = sum_c x, q = sum_c x^2 --------
        for (int idx = tid; idx < REG * REG; idx += 128) {
            const int r = idx / REG, c = idx - r * REG;
            const int l = r * TSTRIDE + c;
            float sp = 0.f, qp = 0.f, sm = 0.f, qm = 0.f;
#pragma unroll
            for (int ch = 0; ch < NCH; ++ch) {
                const float a = raw[cb][ch][l];
                const float b = raw[cb][NCH + ch][l];
                sp += a; qp += a * a;
                sm += b; qm += b * b;
            }
            tP[l] = sp; qP[l] = qp;
            tM[l] = sm; qM[l] = qm;
        }
        __syncthreads();

        // -------- horizontal 7-tap box sums; rows 38/39 zeroed --------
        for (int idx = tid; idx < HROWS * TILE; idx += 128) {
            const int j = idx / TILE, c = idx - j * TILE;
            float a = 0.f, b = 0.f, e = 0.f, f = 0.f;
            if (j < REG) {
                const int base = j * TSTRIDE + c;
#pragma unroll
                for (int d = 0; d < 7; ++d) {
                    a += tP[base + d]; b += qP[base + d];
                    e += tM[base + d]; f += qM[base + d];
                }
            }
            hsP[j * HSTRIDE + c] = a;  hqP[j * HSTRIDE + c] = b;
            hsM[j * HSTRIDE + c] = e;  hqM[j * HSTRIDE + c] = f;
        }
        __syncthreads();

        // -------- vertical 7-tap as banded matmul via V_WMMA_F32_16X16X4_F32 --------
        // D[m][n] = sum_j A[m][j] * H[r0+j][c0+n],  A[m][j] = (m <= j <= m+6)
        v8f accSP = {}, accQP = {}, accSM = {}, accQM = {};
        for (int k0 = 0; k0 < 22; k0 += 4) {        // K chunks: 0,4,8,12,16,20
            const int j0 = k0 + 2 * hi;             // A: VGPR0 -> K=2*hi, VGPR1 -> K=2*hi+1
            const int j1 = j0 + 1;
            v2f a;
            a.x = (j0 >= m && j0 <= m + 6) ? 1.0f : 0.0f;
            a.y = (j1 >= m && j1 <= m + 6) ? 1.0f : 0.0f;

            const int row0 = (r0 + j0) * HSTRIDE + c0 + m;
            const int row1 = (r0 + j1) * HSTRIDE + c0 + m;

            v2f b;
            b.x = hsP[row0]; b.y = hsP[row1];
            accSP = __builtin_amdgcn_wmma_f32_16x16x4_f32(false, a, false, b, (short)0, accSP, false, false);
            b.x = hqP[row0]; b.y = hqP[row1];
            accQP = __builtin_amdgcn_wmma_f32_16x16x4_f32(false, a, false, b, (short)0, accQP, false, false);
            b.x = hsM[row0]; b.y = hsM[row1];
            accSM = __builtin_amdgcn_wmma_f32_16x16x4_f32(false, a, false, b, (short)0, accSM, false, false);
            b.x = hqM[row0]; b.y = hqM[row1];
            accQM = __builtin_amdgcn_wmma_f32_16x16x4_f32(false, a, false, b, (short)0, accQM, false, false);
        }

        // -------- std + smooth-L1, block reduction, per-tile partial --------
        const float inv = 1.0f / NDIVF;
        float lsum = 0.f;
#pragma unroll
        for (int v = 0; v < 8; ++v) {
            const float mup = accSP[v] * inv;
            const float vp  = accQP[v] * inv - mup * mup;
            const float sdp = sqrtf(vp + 1e-8f);
            const float mum = accSM[v] * inv;
            const float vm  = accQM[v] * inv - mum * mum;
            const float sdm = sqrtf(vm + 1e-8f);
            const float d  = sdp - sdm;
            const float ad = fabsf(d);
            lsum += (ad < 1.0f) ? 0.5f * d * d : (ad - 0.5f);
        }
        red[tid] = lsum;
        __syncthreads();
#pragma unroll
        for (int s = 64; s > 0; s >>= 1) {
            if (tid < s) red[tid] += red[tid + s];
            __syncthreads();
        }
        if (tid == 0) partial[baseTile + i] = red[0];
        // next iteration's WAIT_ASYNC + __syncthreads protects red[] and raw[] reuse
    }
}

__global__ __launch_bounds__(256) void final_reduce_kernel(
    const float* __restrict__ partial, float* __restrict__ out)
{
    __shared__ float red[256];
    float s = 0.f;
    for (int i = threadIdx.x; i < NTILES; i += 256) s += partial[i];
    red[threadIdx.x] = s;
    __syncthreads();
#pragma unroll
    for (int st = 128; st > 0; st >>= 1) {
        if (threadIdx.x < st) red[threadIdx.x] += red[threadIdx.x + st];
        __syncthreads();
    }
    if (threadIdx.x == 0) out[0] = red[0] * (1.0f / NTOTAL);
}

extern "C" void kernel_launch(void* const* d_in, const int* in_sizes, int n_in,
                              void* d_out, int out_size, void* d_ws, size_t ws_size,
                              hipStream_t stream)
{
    const float* pred = (const float*)d_in[0];   // pred_moire
    const float* tgt  = (const float*)d_in[1];   // moire
    float* partial = (float*)d_ws;               // 4096 floats of scratch
    float* out = (float*)d_out;

    local_std_smoothl1_kernel<<<dim3(NBLOCKS), 128, 0, stream>>>(pred, tgt, partial);
    final_reduce_kernel<<<1, 256, 0, stream>>>(partial, out);
}